// GASPairCopula_36876589204242
// MI455X (gfx1250) — compile-verified
//
#include <hip/hip_runtime.h>

#define EPS_F 1e-9f

// ---------------------------------------------------------------------------
// Tail-stable float ndtri: ndtri(p) = sqrt(2) * erfinv(2p - 1).
// w = -log(4 p (1-p)) == -log(1 - x^2) computed from p directly so the
// p ~ 1e-9 clip boundary does not cancel to log(0). (Giles erfinv polys.)
// ---------------------------------------------------------------------------
__device__ __forceinline__ float ndtri_fast(float p) {
    float x = fmaf(2.0f, p, -1.0f);
    float w = -__logf(4.0f * p * (1.0f - p));
    float s;
    if (w < 5.0f) {
        w = w - 2.5f;
        s = 2.81022636e-08f;
        s = fmaf(s, w, 3.43273939e-07f);
        s = fmaf(s, w, -3.5233877e-06f);
        s = fmaf(s, w, -4.39150654e-06f);
        s = fmaf(s, w, 0.00021858087f);
        s = fmaf(s, w, -0.00125372503f);
        s = fmaf(s, w, -0.00417768164f);
        s = fmaf(s, w, 0.246640727f);
        s = fmaf(s, w, 1.50140941f);
    } else {
        w = __builtin_sqrtf(w) - 3.0f;
        s = -0.000200214257f;
        s = fmaf(s, w, 0.000100950558f);
        s = fmaf(s, w, 0.00134934322f);
        s = fmaf(s, w, -0.00367342844f);
        s = fmaf(s, w, 0.00573950773f);
        s = fmaf(s, w, -0.0076224613f);
        s = fmaf(s, w, 0.00943887047f);
        s = fmaf(s, w, 1.00167406f);
        s = fmaf(s, w, 2.83297682f);
    }
    return 1.41421356237f * s * x;
}

__device__ __forceinline__ float clip01(float p) {
    return fminf(fmaxf(p, EPS_F), 1.0f - EPS_F);
}

// (sq, xyp) = (x^2 + y^2, x*y) from clipped uniforms.
__device__ __forceinline__ void quantile_pair(float up, float vp, float& sq,
                                              float& xyp) {
    float x = ndtri_fast(clip01(up));
    float y = ndtri_fast(clip01(vp));
    sq = fmaf(x, x, y * y);
    xyp = x * y;
}

// Constant-lane wave32 broadcast via v_readlane_b32: result lands in an SGPR,
// which lets clang keep the uniform recurrence on the SALU-float pipe.
__device__ __forceinline__ float bcast_lane(float v, int lane) {
    return __int_as_float(__builtin_amdgcn_readlane(__float_as_int(v), lane));
}

// Hardware tanh if this toolchain exposes gfx1250's v_tanh_f32; otherwise a
// short exp/rcp chain.
__device__ __forceinline__ float tanh_chain(float x) {
#if __has_builtin(__builtin_amdgcn_tanhf)
    return __builtin_amdgcn_tanhf(x);
#else
    float e2 = __expf(2.0f * x);
    return fmaf(-2.0f, __builtin_amdgcn_rcpf(e2 + 1.0f), 1.0f);
#endif
}

// ---------------------------------------------------------------------------
// Phase 1: embarrassingly parallel. Computes the quantiles AND the two
// sufficient statistics the scan consumes, stored interleaved as
// float2(sq, xyp) so the scan wave reads one contiguous 256B line per
// 32-step chunk and does zero arithmetic on the serial wave's behalf.
// ---------------------------------------------------------------------------
__global__ void stats_precompute_kernel(const float* __restrict__ u,
                                        const float* __restrict__ v,
                                        float2* __restrict__ st, int n) {
    int i = blockIdx.x * blockDim.x + threadIdx.x;
    int stride = gridDim.x * blockDim.x;
    for (; i < n; i += stride) {
        float sq, xyp;
        quantile_pair(u[i], v[i], sq, xyp);
        st[i] = make_float2(sq, xyp);
    }
}

// ---------------------------------------------------------------------------
// Phase 2: single wave32 runs the sequential GAS(1,1) recurrence.
// Per step: two constant-lane v_readlane broadcasts feed the wave-uniform
// chain (lowered to SALU-float by the compiler; trans ops stay on VALU).
// Next chunk's loads issue before the 32 dependent steps; global_prefetch_b8
// pulls 8 chunks (2KB) ahead.
// ---------------------------------------------------------------------------
template <bool FUSED>
__global__ __launch_bounds__(32) void gas_scan_kernel(
    const float2* __restrict__ st, const float* __restrict__ u,
    const float* __restrict__ v, const float* __restrict__ omega_p,
    const float* __restrict__ A_p, const float* __restrict__ Bl_p,
    float* __restrict__ out, int n) {
    const int lane = (int)threadIdx.x;
    const float omega = *omega_p;
    const float A = *A_p;
    const float B = 1.0f / (1.0f + __expf(-*Bl_p));  // sigmoid(B_logit)
    const float c0 = omega * (1.0f - B);             // f' = B*f + A*scaled + c0

    float f = omega;
    float s_var = 1.0f;
    float sum = 0.0f, comp = 0.0f;

    auto step = [&](float sq, float xyp) {
        float th = tanh_chain(f);
        float rho = 0.999f * th;
        float D = fmaf(-rho, rho, 1.0f) + 1e-8f;
        float invD = __builtin_amdgcn_rcpf(D);
        float m2xyp = -2.0f * xyp;                 // off-chain
        float z = fmaf(rho, m2xyp, sq);            // overlaps the rcp
        float ll = fmaf(-0.5f, __logf(D), -0.5f * fmaf(z, invD, -sq));
        float dll = fmaf(rho + xyp, invD, -(rho * z) * (invD * invD));
        float score = dll * (0.999f * fmaf(-th, th, 1.0f));
        s_var = fmaf(0.99f, s_var, 0.01f * (score * score));
        float scaled = score * __builtin_amdgcn_rsqf(s_var + 1e-8f);
        f = fmaf(B, f, fmaf(A, scaled, c0));
        // Kahan accumulation (3 dependent FADDs, well under the f-chain depth)
        float yk = ll - comp;
        float tk = sum + yk;
        comp = (tk - sum) - yk;
        sum = tk;
    };

    // Per-lane fetch of element i -> (sq, xyp); independent of the f-chain.
    auto load_pair = [&](int i, float& sqo, float& xypo) {
        if constexpr (FUSED) {
            quantile_pair(u[i], v[i], sqo, xypo);
        } else {
            float2 p = st[i];
            sqo = p.x;
            xypo = p.y;
        }
    };

    const int full = n >> 5;  // full 32-step chunks

    float sql = 0.0f, xypl = 0.0f;
    if (full > 0) load_pair(lane, sql, xypl);

    for (int c = 0; c < full; ++c) {
        float sqn = 0.0f, xypn = 0.0f;
        if (c + 1 < full) load_pair((c + 1) * 32 + lane, sqn, xypn);

        // gfx1250 global_prefetch_b8: pull 8 chunks (2KB) ahead toward WGP$
        int pf = (c + 8) * 32 + lane;
        if (pf < n) {
            if constexpr (FUSED) {
                __builtin_prefetch((const void*)(u + pf), 0, 3);
                __builtin_prefetch((const void*)(v + pf), 0, 3);
            } else {
                __builtin_prefetch((const void*)(st + pf), 0, 3);
            }
        }

#pragma unroll
        for (int k = 0; k < 32; ++k) {
            step(bcast_lane(sql, k), bcast_lane(xypl, k));
        }
        sql = sqn;
        xypl = xypn;
    }

    // tail (n is 2^20 in practice, so this is normally empty)
    for (int i = full << 5; i < n; ++i) {
        float sq, xyp;
        load_pair(i, sq, xyp);
        step(sq, xyp);
    }

    if (lane == 0) out[0] = sum;
}

// ---------------------------------------------------------------------------
// Host launcher. Inputs (setup_inputs order): u_data, v_data, omega, A,
// B_logit. Output: single f32 scalar. Uses d_ws for the 8MB float2 statistics
// array when it fits; otherwise runs the fused fallback. Deterministic,
// stream-only, graph-capture safe.
// ---------------------------------------------------------------------------
extern "C" void kernel_launch(void* const* d_in, const int* in_sizes, int n_in,
                              void* d_out, int out_size, void* d_ws,
                              size_t ws_size, hipStream_t stream) {
    const float* u = (const float*)d_in[0];
    const float* v = (const float*)d_in[1];
    const float* omega = (const float*)d_in[2];
    const float* A = (const float*)d_in[3];
    const float* Bl = (const float*)d_in[4];
    float* out = (float*)d_out;
    const int n = in_sizes[0];

    const bool two_phase =
        (d_ws != nullptr) && (ws_size >= (size_t)n * sizeof(float2));

    if (two_phase) {
        float2* st = (float2*)d_ws;
        int threads = 256;
        int blocks = (n + threads - 1) / threads;
        if (blocks > 8192) blocks = 8192;
        stats_precompute_kernel<<<blocks, threads, 0, stream>>>(u, v, st, n);
        gas_scan_kernel<false><<<1, 32, 0, stream>>>(st, u, v, omega, A, Bl,
                                                     out, n);
    } else {
        gas_scan_kernel<true><<<1, 32, 0, stream>>>(nullptr, u, v, omega, A,
                                                    Bl, out, n);
    }
}